// StrGNN_7327214207522
// MI455X (gfx1250) — compile-verified
//
#include <hip/hip_runtime.h>
#include <hip/hip_bf16.h>

#define T_SNAP 4
#define EQ     4096
#define NN     8192
#define HD     32
#define KHOPS  2
#define EQ3    (EQ * 3)

typedef __attribute__((ext_vector_type(2))) float v2f;
typedef __attribute__((ext_vector_type(8))) float v8f;

__device__ __forceinline__ v8f wmma_f32(v2f a, v2f b, v8f c) {
  // D = A(16x4,f32) * B(4x16,f32) + C(16x16,f32)
  return __builtin_amdgcn_wmma_f32_16x16x4_f32(false, a, false, b, (short)0, c,
                                               false, false);
}

__device__ __forceinline__ void push3(float c, float& t0, float& t1, float& t2) {
  if (c > t0)      { t2 = t1; t1 = t0; t0 = c; }
  else if (c > t1) { t2 = t1; t1 = c; }
  else if (c > t2) { t2 = c; }
}

// ---------------------------------------------------------------------------
// Kernel A: counting-sort each snapshot's edges by dst (order within a run is
// irrelevant: we only need per-dst run lengths later). One block per graph.
// ---------------------------------------------------------------------------
__global__ __launch_bounds__(1024) void sort_edges_kernel(
    const int* __restrict__ ei, int* __restrict__ sS, int* __restrict__ sD) {
  __shared__ unsigned hist[NN];     // 32KB
  __shared__ unsigned partial[1024];
  const int t   = blockIdx.x;
  const int tid = threadIdx.x;
  const int* src = ei + (size_t)t * 2 * EQ;
  const int* dst = src + EQ;

  for (int i = tid; i < NN; i += 1024) hist[i] = 0u;
  __syncthreads();
  for (int e = tid; e < EQ; e += 1024) atomicAdd(&hist[dst[e]], 1u);
  __syncthreads();

  // block-wide exclusive scan of 8192 bins (8 bins/thread + scan of partials)
  unsigned c[8], s = 0;
  const int base = tid * 8;
#pragma unroll
  for (int k = 0; k < 8; ++k) { c[k] = hist[base + k]; s += c[k]; }
  partial[tid] = s;
  __syncthreads();
  for (int off = 1; off < 1024; off <<= 1) {
    unsigned v = (tid >= off) ? partial[tid - off] : 0u;
    __syncthreads();
    partial[tid] += v;
    __syncthreads();
  }
  unsigned run = partial[tid] - s;  // exclusive prefix
#pragma unroll
  for (int k = 0; k < 8; ++k) { hist[base + k] = run; run += c[k]; }
  __syncthreads();

  for (int e = tid; e < EQ; e += 1024) {
    const int d = dst[e];
    const unsigned p = atomicAdd(&hist[d], 1u);
    sS[(size_t)t * EQ + p] = src[e];
    sD[(size_t)t * EQ + p] = d;
  }
}

// ---------------------------------------------------------------------------
// Kernel B: per (graph, 32-query block): bit-packed 2-hop reachability in LDS
// (u64: low 32 bits = Ru bits, high 32 = Rv bits), edge-inclusion mask, then
// per-query top-3 in-degree via run-length scan over dst-sorted edges.
// Dynamic LDS: 2*64KB reach buffers + edges + incl + partial tops (~179KB).
// ---------------------------------------------------------------------------
__global__ __launch_bounds__(256) void reach_top3_kernel(
    const int* __restrict__ ei, const int* __restrict__ sS,
    const int* __restrict__ sD, float* __restrict__ tops) {
  extern __shared__ unsigned long long smemB[];
  unsigned long long* RA = smemB;            // NN u64
  unsigned long long* RB = RA + NN;          // NN u64
  int*      eS   = (int*)(RB + NN);          // EQ
  int*      eD   = eS + EQ;                  // EQ
  unsigned* incl = (unsigned*)(eD + EQ);     // EQ
  float*    ptop = (float*)(incl + EQ);      // 8*32*3

  const int qb  = blockIdx.x;   // query block (32 queries)
  const int t   = blockIdx.y;   // snapshot
  const int tid = threadIdx.x;
  const int* src = ei + (size_t)t * 2 * EQ;
  const int* dst = src + EQ;

  for (int e = tid; e < EQ; e += 256) { eS[e] = src[e]; eD[e] = dst[e]; }
  for (int i = tid; i < NN; i += 256) RA[i] = 0ull;
  __syncthreads();

  if (tid < 32) {  // seeds: uq/vq from last snapshot's edge list
    const int q = qb * 32 + tid;
    const int* uq = ei + (size_t)(T_SNAP - 1) * 2 * EQ;
    const int* vq = uq + EQ;
    atomicOr(&RA[uq[q]], 1ull << tid);
    atomicOr(&RA[vq[q]], 1ull << (32 + tid));
  }
  __syncthreads();

  unsigned long long* Rold = RA;
  unsigned long long* Rnew = RB;
  for (int hop = 0; hop < KHOPS; ++hop) {
    for (int i = tid; i < NN; i += 256) Rnew[i] = Rold[i];
    __syncthreads();
    for (int e = tid; e < EQ; e += 256) {
      const unsigned long long w = Rold[eD[e]];
      if (w) atomicOr(&Rnew[eS[e]], w);
    }
    __syncthreads();
    unsigned long long* tmp = Rold; Rold = Rnew; Rnew = tmp;
  }

  // reload edges in dst-sorted order (overwrite), then inclusion bits
  for (int e = tid; e < EQ; e += 256) {
    eS[e] = sS[(size_t)t * EQ + e];
    eD[e] = sD[(size_t)t * EQ + e];
  }
  __syncthreads();
  for (int e = tid; e < EQ; e += 256) {
    const unsigned long long a = Rold[eS[e]] & Rold[eD[e]];
    incl[e] = (unsigned)a | (unsigned)(a >> 32);  // Ru&Ru | Rv&Rv per query
  }
  __syncthreads();

  // run-length top-3: each wave scans a run-boundary-aligned chunk, lane=query
  const int wave = tid >> 5, lane = tid & 31;
  {
    int s0 = wave * (EQ / 8);
    int e0 = s0 + (EQ / 8);
    if (wave > 0) while (s0 < EQ && eD[s0] == eD[s0 - 1]) ++s0;
    while (e0 < EQ && eD[e0] == eD[e0 - 1]) ++e0;
    float t0 = 0.f, t1 = 0.f, t2 = 0.f;
    int curD = -1; unsigned cnt = 0;
    for (int i = s0; i < e0; ++i) {
      const int d = eD[i];
      if (d != curD) { push3((float)cnt, t0, t1, t2); curD = d; cnt = 0; }
      cnt += (incl[i] >> lane) & 1u;
    }
    push3((float)cnt, t0, t1, t2);
    ptop[(wave * 32 + lane) * 3 + 0] = t0;
    ptop[(wave * 32 + lane) * 3 + 1] = t1;
    ptop[(wave * 32 + lane) * 3 + 2] = t2;
  }
  __syncthreads();
  if (tid < 32) {
    float t0 = 0.f, t1 = 0.f, t2 = 0.f;
    for (int w = 0; w < 8; ++w)
#pragma unroll
      for (int k = 0; k < 3; ++k) push3(ptop[(w * 32 + tid) * 3 + k], t0, t1, t2);
    const int q = qb * 32 + tid;
    float* o = tops + (size_t)t * EQ3 + (size_t)q * 3;
    o[0] = t0; o[1] = t1; o[2] = t2;
  }
}

// ---------------------------------------------------------------------------
// Kernel C: GCN feature + 4-step GRU with f32 WMMA (16x16x4).
// Wave = 16 batch rows; 6 gate tiles x 8 K-steps x {W_ih,W_hh} = 96 WMMA/step.
// r/z gate tiles accumulate gi+gh jointly; n-gate keeps ig/hg separate.
// ---------------------------------------------------------------------------
__global__ __launch_bounds__(128) void gru_wmma_kernel(
    const float* __restrict__ tops, const float* __restrict__ Wg,
    const float* __restrict__ bg, const float* __restrict__ Wih,
    const float* __restrict__ Whh, const float* __restrict__ bih,
    const float* __restrict__ bhh, float* __restrict__ out) {
  __shared__ float sWih[96 * 32];
  __shared__ float sWhh[96 * 32];
  __shared__ float sbih[96], sbhh[96], swg[32], sbg[32];
  __shared__ float hbuf[4][16][32];  // per-wave hidden state (row-major)

  const int tid = threadIdx.x;
  for (int i = tid; i < 96 * 32; i += 128) { sWih[i] = Wih[i]; sWhh[i] = Whh[i]; }
  if (tid < 96) { sbih[tid] = bih[tid]; sbhh[tid] = bhh[tid]; }
  if (tid < 32) { swg[tid] = Wg[tid]; sbg[tid] = bg[tid]; }

  const int wave = tid >> 5, lane = tid & 31;
  const int n    = lane & 15;        // N index in B/C/D tiles; A-row index
  const int hh2  = (lane >> 4) * 2;  // A/B K sub-offset for lane half
  const int mOff = (lane >> 4) * 8;  // C/D row offset for lane half
  const int rowBase = (blockIdx.x * 4 + wave) * 16;

  for (int i = lane; i < 16 * 32; i += 32) (&hbuf[wave][0][0])[i] = 0.f;
  __syncthreads();

  float hnew[2][8];
  for (int t = 0; t < T_SNAP; ++t) {
    // A-fragments: X (GCN: relu(s*Wg + bg), rank-1 in features) and Hprev
    v2f ax[8], ah[8];
    const float s = tops[(size_t)t * EQ3 + rowBase + n];
#pragma unroll
    for (int k0 = 0; k0 < 8; ++k0) {
      const int kb = k0 * 4 + hh2;
      ax[k0][0] = fmaxf(fmaf(s, swg[kb],     sbg[kb]),     0.f);
      ax[k0][1] = fmaxf(fmaf(s, swg[kb + 1], sbg[kb + 1]), 0.f);
      ah[k0][0] = hbuf[wave][n][kb];
      ah[k0][1] = hbuf[wave][n][kb + 1];
    }

    const v8f vz = {0.f, 0.f, 0.f, 0.f, 0.f, 0.f, 0.f, 0.f};
    v8f acc[8];  // [0..3]=r,z (gi+gh summed); [4,5]=ig; [6,7]=hg
#pragma unroll
    for (int a = 0; a < 8; ++a) acc[a] = vz;

#pragma unroll
    for (int jt = 0; jt < 6; ++jt) {
#pragma unroll
      for (int k0 = 0; k0 < 8; ++k0) {
        const int kb = k0 * 4 + hh2;
        const int j  = jt * 16 + n;  // gate column
        v2f bi, bh;
        bi[0] = sWih[j * 32 + kb]; bi[1] = sWih[j * 32 + kb + 1];
        bh[0] = sWhh[j * 32 + kb]; bh[1] = sWhh[j * 32 + kb + 1];
        acc[jt] = wmma_f32(ax[k0], bi, acc[jt]);
        const int dh = (jt < 4) ? jt : jt + 2;
        acc[dh] = wmma_f32(ah[k0], bh, acc[dh]);
      }
    }

    // gates (D-layout: element (m = r+mOff, col = g*16+n) in acc[*][r])
#pragma unroll
    for (int g = 0; g < 2; ++g) {
      const int jc = g * 16 + n;
      const float brz_r = sbih[jc] + sbhh[jc];
      const float brz_z = sbih[32 + jc] + sbhh[32 + jc];
      const float bg_i  = sbih[64 + jc];
      const float bg_h  = sbhh[64 + jc];
#pragma unroll
      for (int r = 0; r < 8; ++r) {
        const int m = r + mOff;
        const float rg = 1.f / (1.f + __expf(-(acc[g][r] + brz_r)));
        const float zg = 1.f / (1.f + __expf(-(acc[2 + g][r] + brz_z)));
        const float ig = acc[4 + g][r] + bg_i;
        const float hg = acc[6 + g][r] + bg_h;
        const float ng = tanhf(fmaf(rg, hg, ig));
        const float hp = hbuf[wave][m][jc];
        hnew[g][r] = fmaf(zg, hp - ng, ng);  // (1-z)*n + z*hp
      }
    }
    __syncthreads();
#pragma unroll
    for (int g = 0; g < 2; ++g)
#pragma unroll
      for (int r = 0; r < 8; ++r) hbuf[wave][r + mOff][g * 16 + n] = hnew[g][r];
    __syncthreads();
  }

  // final hidden -> out[(q*3+p)*32 + j], coalesced per 16-lane half
#pragma unroll
  for (int g = 0; g < 2; ++g)
#pragma unroll
    for (int r = 0; r < 8; ++r)
      out[(size_t)(rowBase + r + mOff) * 32 + g * 16 + n] = hnew[g][r];
}

// ---------------------------------------------------------------------------
extern "C" void kernel_launch(void* const* d_in, const int* in_sizes, int n_in,
                              void* d_out, int out_size, void* d_ws,
                              size_t ws_size, hipStream_t stream) {
  const int*   ei  = (const int*)d_in[0];
  const float* Wg  = (const float*)d_in[1];
  const float* bg  = (const float*)d_in[2];
  const float* Wih = (const float*)d_in[3];
  const float* Whh = (const float*)d_in[4];
  const float* bih = (const float*)d_in[5];
  const float* bhh = (const float*)d_in[6];
  float*       out = (float*)d_out;

  int*   sS   = (int*)d_ws;                    // T*EQ
  int*   sD   = sS + (size_t)T_SNAP * EQ;      // T*EQ
  float* tops = (float*)(sD + (size_t)T_SNAP * EQ);  // T*EQ*3

  sort_edges_kernel<<<T_SNAP, 1024, 0, stream>>>(ei, sS, sD);

  const size_t smemB = (size_t)NN * 8 * 2      // two u64 reach buffers
                     + (size_t)EQ * 4 * 2      // edge src/dst
                     + (size_t)EQ * 4          // incl bits
                     + 8 * 32 * 3 * 4;         // partial top-3
  dim3 gridB(EQ / 32, T_SNAP);
  reach_top3_kernel<<<gridB, 256, smemB, stream>>>(ei, sS, sD, tops);

  gru_wmma_kernel<<<EQ3 / 64, 128, 0, stream>>>(tops, Wg, bg, Wih, Whh, bih,
                                                bhh, out);
}